// EndPointAggregator_11012296147565
// MI455X (gfx1250) — compile-verified
//
#include <hip/hip_runtime.h>
#include <hip/hip_bf16.h>

#define SEQ_LEN   4096
#define INPUT_DIM 768
#define N_SPANS   200000
#define OUT_DIM   1538   // 768 + 768 + 2

typedef float v2f __attribute__((ext_vector_type(2)));
typedef float v8f __attribute__((ext_vector_type(8)));

// gfx1250 has a native V_TANH_F32 transcendental; use it if the toolchain
// exposes the builtin, otherwise fall back to libm tanhf.
__device__ __forceinline__ float fast_tanh(float x) {
#if __has_builtin(__builtin_amdgcn_tanhf)
    return __builtin_amdgcn_tanhf(x);
#else
    return tanhf(x);
#endif
}

// ---------------------------------------------------------------------------
// Kernel 1: gather start/end embedding rows into the output.
// Store-bandwidth bound (1.23 GB out); emb table (12.6 MB) is L2-resident.
// One block per span; 256 threads; each thread moves 3 float2 (768 float2 =
// 2 rows x 384). B64 accesses because odd output rows are only 8B-aligned.
// Non-temporal stores: output is write-once, keep it out of L2 so the hot
// embedding table stays resident.
// ---------------------------------------------------------------------------
__global__ __launch_bounds__(256) void endpoint_gather_kernel(
    const float* __restrict__ emb,
    const long long* __restrict__ spans,
    float* __restrict__ out)
{
    const int s = blockIdx.x;
    const int t = threadIdx.x;

    // Uniform per block -> compiler scalarizes these loads (s_load).
    const long long st = spans[2 * (long long)s];
    const long long en = spans[2 * (long long)s + 1];

    const float* __restrict__ srow = emb + (size_t)st * INPUT_DIM;
    const float* __restrict__ erow = emb + (size_t)en * INPUT_DIM;
    float* __restrict__ orow = out + (size_t)s * OUT_DIM;

#pragma unroll
    for (int k = 0; k < 3; ++k) {
        const int i   = t + k * 256;          // 0..767 (float2 index, both rows)
        const int row = (i >= 384) ? 1 : 0;   // 0 = start row, 1 = end row
        const int col = i - row * 384;        // float2 column within the row

        const float* __restrict__ src = row ? erow : srow;
        v2f val = *(const v2f*)(src + col * 2);
        __builtin_nontemporal_store(val, (v2f*)(orow + row * INPUT_DIM + col * 2));
    }
}

// ---------------------------------------------------------------------------
// Kernel 2: dist_emb = tanh(dist * w + b) for 2 output columns, computed with
// V_WMMA_F32_16X16X4_F32, 16 spans per wave.
//   A (16x4 f32): lane m (0..15), element .x = dist[base+m] at K=0; all other
//                 K slots zero (lanes 16..31 hold K=2/3 -> zero).
//   B (4x16 f32): K=0 row in VGPR0 lanes 0..15 -> lane0.x = w0, lane1.x = w1.
//   D: column n lives in lane n (M=0..7 in c[0..7]) and lane 16+n (M=8..15).
// ---------------------------------------------------------------------------
__global__ __launch_bounds__(256) void dist_wmma_kernel(
    const long long* __restrict__ spans,
    const float* __restrict__ dist_w,   // shape (2,1) -> w0 = dist_w[0], w1 = dist_w[1]
    const float* __restrict__ dist_b,   // shape (2,)
    float* __restrict__ out)
{
    const int wave = (blockIdx.x * blockDim.x + threadIdx.x) >> 5;
    const int lane = threadIdx.x & 31;
    const int base = wave * 16;
    if (base >= N_SPANS) return;        // wave-uniform exit: EXEC stays all-1s

    // A matrix: dist values in K=0 column.
    v2f a = {0.0f, 0.0f};
    if (lane < 16) {
        const long long st = spans[2 * (long long)(base + lane)];
        const long long en = spans[2 * (long long)(base + lane) + 1];
        a.x = (float)(en - st);
    }

    // B matrix: weights in K=0 row, columns 0 and 1.
    v2f bm = {0.0f, 0.0f};
    if (lane == 0) bm.x = dist_w[0];
    if (lane == 1) bm.x = dist_w[1];

    v8f c = {};
    // EXEC is all-1s here (divergent setup re-converged above).
    c = __builtin_amdgcn_wmma_f32_16x16x4_f32(
            /*neg_a=*/false, a, /*neg_b=*/false, bm,
            /*c_mod=*/(short)0, c, /*reuse_a=*/false, /*reuse_b=*/false);

    // Extract columns n = 0,1: held by lanes 0,1 (M=0..7) and 16,17 (M=8..15).
    const int n = lane & 15;
    if (n < 2) {
        const int   mbase = (lane >> 4) * 8;
        const float bias  = dist_b[n];
#pragma unroll
        for (int i = 0; i < 8; ++i) {
            const int m = mbase + i;
            out[(size_t)(base + m) * OUT_DIM + (INPUT_DIM * 2) + n] =
                fast_tanh(c[i] + bias);
        }
    }
}

// ---------------------------------------------------------------------------
extern "C" void kernel_launch(void* const* d_in, const int* in_sizes, int n_in,
                              void* d_out, int out_size, void* d_ws, size_t ws_size,
                              hipStream_t stream) {
    (void)in_sizes; (void)n_in; (void)out_size; (void)d_ws; (void)ws_size;

    const float*     emb   = (const float*)d_in[0];      // (4096, 768) f32
    const long long* spans = (const long long*)d_in[1];  // (200000, 2) i64
    const float*     dw    = (const float*)d_in[2];      // (2, 1) f32
    const float*     db    = (const float*)d_in[3];      // (2,)  f32
    float*           out   = (float*)d_out;              // (200000, 1538) f32

    // Gather: one block per span.
    endpoint_gather_kernel<<<N_SPANS, 256, 0, stream>>>(emb, spans, out);

    // dist_emb via WMMA: 16 spans per wave, 8 waves per block.
    const int nwaves  = (N_SPANS + 15) / 16;   // 12500
    const int nblocks = (nwaves + 7) / 8;      // 1563
    dist_wmma_kernel<<<nblocks, 256, 0, stream>>>(spans, dw, db, out);
}